// DifferentiableBiquad_3908420239557
// MI455X (gfx1250) — compile-verified
//
#include <hip/hip_runtime.h>
#include <stdint.h>

// ---------------------------------------------------------------------------
// Biquad (DF2T) over (B=64, N=262144) as a chunk-parallel linear scan.
//   Pass1: per-chunk forced end-state, TDM (tensor_load_to_lds) staged tiles.
//   Pass2: per-channel combine = wave-parallel affine scan (M = A^CLEN const).
//   Pass3: state-space block processing: 5x V_WMMA_F32_16X16X4_F32 per
//          16 timesteps x 16 chunks; TDM double-buffered X in, async LDS->
//          global Y out.
// Roofline: 64MB x-read (HBM) + 64MB y-write; pass3 x re-read hits 192MB L2.
// ---------------------------------------------------------------------------

#define NCH     64
#define NSAMP   262144
#define CHUNKS  256                  // chunks per channel
#define CLEN    (NSAMP / CHUNKS)     // 1024 samples per chunk
#define TBLK    16                   // timesteps per block
#define NBLK    (CLEN / TBLK)        // 64 blocks per chunk

typedef float v2f __attribute__((ext_vector_type(2)));
typedef float v8f __attribute__((ext_vector_type(8)));
typedef unsigned int u32x4 __attribute__((ext_vector_type(4)));
typedef int   i32x4 __attribute__((ext_vector_type(4)));
typedef int   i32x8 __attribute__((ext_vector_type(8)));
typedef __attribute__((address_space(3))) int lds_i32;   // LDS (local) int
typedef __attribute__((address_space(1))) int glb_i32;   // global int

#if __has_builtin(__builtin_amdgcn_wmma_f32_16x16x4_f32)
#define HAVE_WMMA 1
#else
#define HAVE_WMMA 0
#endif

#if __has_builtin(__builtin_amdgcn_tensor_load_to_lds) && __has_builtin(__builtin_amdgcn_s_wait_tensorcnt)
#define HAVE_TDM 1
#else
#define HAVE_TDM 0
#endif

#if __has_builtin(__builtin_amdgcn_global_store_async_from_lds_b32) && __has_builtin(__builtin_amdgcn_s_wait_asynccnt)
#define HAVE_ASYNC_ST 1
#else
#define HAVE_ASYNC_ST 0
#endif

__device__ __forceinline__ void wave_sync() {
#if __has_builtin(__builtin_amdgcn_wave_barrier)
  __builtin_amdgcn_wave_barrier();
#else
  __syncthreads();
#endif
}

#if HAVE_TDM
// 2D TDM tile: tile_dim0 = 16 dwords (contiguous time), tile_dim1 = nrows
// (chunks), memory row stride = CLEN elements. LDS gets +1 dword pad per 16
// dwords (pad_interval code 3, pad_amount 0) -> conflict-free stride-17 rows.
__device__ __forceinline__ void tdm_load_tile16(uint32_t lds_byte, uint64_t gaddr,
                                                int nrows) {
  uint32_t glo = (uint32_t)__builtin_amdgcn_readfirstlane((int)(uint32_t)gaddr);
  uint32_t ghi = (uint32_t)__builtin_amdgcn_readfirstlane((int)(uint32_t)(gaddr >> 32));
  uint32_t lds = (uint32_t)__builtin_amdgcn_readfirstlane((int)lds_byte);
  u32x4 g0;
  g0.x = 1u;                                   // D# group0: count=1 (valid)
  g0.y = lds;                                  // lds_addr (bytes)
  g0.z = glo;                                  // global_addr[31:0]
  g0.w = (ghi & 0x01ffffffu) | (2u << 30);     // global_addr[56:32] | type=2
  i32x8 g1;
  g1[0] = (2 << 16) | (1 << 20) | (3 << 22);   // data_size=4B, pad_en, 16dw, +1dw
  g1[1] = (int)0xffff0000u;                    // tensor_dim0 = 0xffff (no OOB)
  g1[2] = (int)0xffff0000u;                    // tensor_dim1 = 0xffff
  g1[3] = (TBLK << 16);                        // tile_dim0 = 16
  g1[4] = nrows;                               // tile_dim1 ; tile_dim2 = 0
  g1[5] = CLEN;                                // tensor_dim0_stride (elements)
  g1[6] = 0;
  g1[7] = 0;
  i32x4 z4 = {0, 0, 0, 0};
#if __clang_major__ >= 23
  i32x8 z8 = {0, 0, 0, 0, 0, 0, 0, 0};
  __builtin_amdgcn_tensor_load_to_lds(g0, g1, z4, z4, z8, 0);
#else
  __builtin_amdgcn_tensor_load_to_lds(g0, g1, z4, z4, 0);
#endif
}
#endif

// ---------------------------------------------------------------------------
// Pass 1: one thread per chunk (wave = 32 chunks); forced end-state only.
// ---------------------------------------------------------------------------
__global__ __launch_bounds__(256) void biquad_pass1(
    const float* __restrict__ x,
    const float* __restrict__ b0v, const float* __restrict__ b1v,
    const float* __restrict__ b2v, const float* __restrict__ a1v,
    const float* __restrict__ a2v, float* __restrict__ st)
{
  __shared__ float tile[8][2][32 * 17];        // per-wave double buffer
  const int c    = blockIdx.x;
  const int tid  = threadIdx.x;
  const int wave = tid >> 5;
  const int lane = tid & 31;

  const float B0 = b0v[c], B1 = b1v[c], B2 = b2v[c];
  const float A1 = a1v[c], A2 = a2v[c];
  const float* __restrict__ xc = x + (size_t)c * NSAMP;
  const int cbase = wave * 32;                 // first chunk of this wave

  float s1 = 0.0f, s2 = 0.0f;
#if HAVE_TDM
  tdm_load_tile16((uint32_t)(uintptr_t)&tile[wave][0][0],
                  (uint64_t)(uintptr_t)(xc + (size_t)cbase * CLEN), 32);
#endif
  for (int ti = 0; ti < NBLK; ++ti) {
    const int cur = ti & 1;
    float* tw = &tile[wave][cur][0];
#if HAVE_TDM
    if (ti + 1 < NBLK) {
      asm volatile("s_wait_dscnt 0x0" ::: "memory");   // WAR vs in-flight ds reads
      tdm_load_tile16((uint32_t)(uintptr_t)&tile[wave][cur ^ 1][0],
                      (uint64_t)(uintptr_t)(xc + (size_t)cbase * CLEN + (ti + 1) * TBLK),
                      32);
      __builtin_amdgcn_s_wait_tensorcnt(1);    // current tile done, prefetch in flight
    } else {
      __builtin_amdgcn_s_wait_tensorcnt(0);
    }
    wave_sync();
#else
    #pragma unroll
    for (int k = 0; k < 16; ++k) {             // 512 floats/wave, 64B runs
      int flat = lane + 32 * k;
      int n = flat >> 4, j = flat & 15;
      tw[n * 17 + j] = xc[(size_t)(cbase + n) * CLEN + ti * TBLK + j];
    }
    if (ti + 1 < NBLK)
      __builtin_prefetch(&xc[(size_t)(cbase + lane) * CLEN + (ti + 1) * TBLK], 0, 1);
    wave_sync();
#endif
    #pragma unroll
    for (int t = 0; t < TBLK; ++t) {           // banks 17*lane+t: conflict-free
      float xt  = tw[lane * 17 + t];
      float out = fmaf(B0, xt, s1);
      s1 = fmaf(B1, xt, fmaf(-A1, out, s2));
      s2 = fmaf(B2, xt, -A2 * out);
    }
    wave_sync();
  }
  float* p = st + ((size_t)c * CHUNKS + cbase + lane) * 2;
  p[0] = s1;
  p[1] = s2;
}

// ---------------------------------------------------------------------------
// Pass 2: wave-parallel affine scan per channel.
//   chunk map: s' = M*s + v_j, M = A^CLEN (constant per channel).
//   lane t folds 8 chunks -> (M^8, u_t); Kogge-Stone composes 32 segments.
// ---------------------------------------------------------------------------
__global__ __launch_bounds__(256) void biquad_pass2(
    const float* __restrict__ a1v, const float* __restrict__ a2v,
    float* __restrict__ st)
{
  const int gid  = blockIdx.x * blockDim.x + threadIdx.x;
  const int c    = gid >> 5;                   // one wave per channel
  const int lane = threadIdx.x & 31;
  if (c >= NCH) return;
  const float A1 = a1v[c], A2 = a2v[c];

  float m00 = -A1, m01 = 1.0f, m10 = -A2, m11 = 0.0f;   // A
  #pragma unroll
  for (int i = 0; i < 10; ++i) {               // M = A^1024
    float t00 = fmaf(m00, m00, m01 * m10);
    float t01 = fmaf(m00, m01, m01 * m11);
    float t10 = fmaf(m10, m00, m11 * m10);
    float t11 = fmaf(m10, m01, m11 * m11);
    m00 = t00; m01 = t01; m10 = t10; m11 = t11;
  }

  float* p = st + ((size_t)c * CHUNKS + lane * 8) * 2;
  float v1[8], v2[8];
  #pragma unroll
  for (int i = 0; i < 8; ++i) { v1[i] = p[2 * i]; v2[i] = p[2 * i + 1]; }

  float u1 = 0.0f, u2 = 0.0f;                  // forced term of lane's segment
  #pragma unroll
  for (int i = 0; i < 8; ++i) {
    float n1 = fmaf(m00, u1, fmaf(m01, u2, v1[i]));
    float n2 = fmaf(m10, u1, fmaf(m11, u2, v2[i]));
    u1 = n1; u2 = n2;
  }

  float P00 = m00, P01 = m01, P10 = m10, P11 = m11;     // -> M^8
  #pragma unroll
  for (int i = 0; i < 3; ++i) {
    float t00 = fmaf(P00, P00, P01 * P10);
    float t01 = fmaf(P00, P01, P01 * P11);
    float t10 = fmaf(P10, P00, P11 * P10);
    float t11 = fmaf(P10, P01, P11 * P11);
    P00 = t00; P01 = t01; P10 = t10; P11 = t11;
  }

  float q1 = u1, q2 = u2;                      // inclusive scan of (P, q)
  #pragma unroll
  for (int d = 1; d < 32; d <<= 1) {
    float o00 = __shfl_up(P00, d), o01 = __shfl_up(P01, d);
    float o10 = __shfl_up(P10, d), o11 = __shfl_up(P11, d);
    float oq1 = __shfl_up(q1, d),  oq2 = __shfl_up(q2, d);
    if (lane >= d) {
      q1 = fmaf(P00, oq1, fmaf(P01, oq2, q1));
      q2 = fmaf(P10, oq1, fmaf(P11, oq2, q2));
      float t00 = fmaf(P00, o00, P01 * o10);
      float t01 = fmaf(P00, o01, P01 * o11);
      float t10 = fmaf(P10, o00, P11 * o10);
      float t11 = fmaf(P10, o01, P11 * o11);
      P00 = t00; P01 = t01; P10 = t10; P11 = t11;
    }
  }
  float s1 = __shfl_up(q1, 1);                 // exclusive: segment-entry state
  float s2 = __shfl_up(q2, 1);
  if (lane == 0) { s1 = 0.0f; s2 = 0.0f; }

  #pragma unroll
  for (int i = 0; i < 8; ++i) {                // rewrite st[] with entry states
    p[2 * i]     = s1;
    p[2 * i + 1] = s2;
    float n1 = fmaf(m00, s1, fmaf(m01, s2, v1[i]));
    float n2 = fmaf(m10, s1, fmaf(m11, s2, v2[i]));
    s1 = n1; s2 = n2;
  }
}

// ---------------------------------------------------------------------------
// Pass 3: WMMA state-space block processing.
//   A (per channel, constant): 16x20 = [h-Toeplitz | g1 g2 | 0 0]
//   B (per block):             20x16 = [X(16t x 16 chunks); s1; s2; 0; 0]
//   D = 16x16: row = time-in-block, col = chunk.
// ---------------------------------------------------------------------------
__device__ __forceinline__ float aval(int r, int col, const float* hg) {
  if (col < 16)  return (r >= col) ? hg[r - col] : 0.0f;   // h[i-j]
  if (col == 16) return hg[16 + r];                        // g1
  if (col == 17) return hg[32 + r];                        // g2
  return 0.0f;                                             // K padding
}

__global__ __launch_bounds__(256) void biquad_pass3(
    const float* __restrict__ x,
    const float* __restrict__ b0v, const float* __restrict__ b1v,
    const float* __restrict__ b2v, const float* __restrict__ a1v,
    const float* __restrict__ a2v,
    const float* __restrict__ st, float* __restrict__ y)
{
  __shared__ float hg[48];                     // h[16], g1[16], g2[16]
  __shared__ float xbuf[8][2][16 * 17];        // per-wave double-buffered X tile
  __shared__ float ybuf[8][16 * 17];           // per-wave Y tile
  __shared__ float s1s[8][32];                 // lanes 16..31 stay zero
  __shared__ float s2s[8][32];

  const int c    = blockIdx.x >> 1;
  const int got  = blockIdx.x & 1;
  const int tid  = threadIdx.x;
  const int wave = tid >> 5;
  const int lane = tid & 31;
  const int r    = lane & 15;
  const int hi   = lane >> 4;

  const float B0 = b0v[c], B1 = b1v[c], B2 = b2v[c];
  const float A1 = a1v[c], A2 = a2v[c];

  if (tid == 0) {
    hg[0] = B0;
    hg[1] = fmaf(-A1, hg[0], B1);
    hg[2] = fmaf(-A1, hg[1], fmaf(-A2, hg[0], B2));
    for (int k = 3; k < 16; ++k) hg[k] = fmaf(-A1, hg[k - 1], -A2 * hg[k - 2]);
    hg[16] = 1.0f; hg[17] = -A1;               // homogeneous response to s1=1
    for (int k = 2; k < 16; ++k) hg[16 + k] = fmaf(-A1, hg[16 + k - 1], -A2 * hg[16 + k - 2]);
    hg[32] = 0.0f; hg[33] = 1.0f;              // homogeneous response to s2=1
    for (int k = 2; k < 16; ++k) hg[32 + k] = fmaf(-A1, hg[32 + k - 1], -A2 * hg[32 + k - 2]);
  }
  __syncthreads();

  v2f afrag[5];
  #pragma unroll
  for (int kk = 0; kk < 5; ++kk) {             // A 16x4 frag: v0=K(4kk+2*hi), v1=+1
    int c0 = 4 * kk + 2 * hi;
    afrag[kk].x = aval(r, c0,     hg);
    afrag[kk].y = aval(r, c0 + 1, hg);
  }

  const int base = got * 128 + wave * 16;      // this wave's first chunk
  const float* __restrict__ xc = x + (size_t)c * NSAMP;
  float* __restrict__ yc = y + (size_t)c * NSAMP;

  {                                            // states, upper 16 lanes = 0
    float v1 = 0.0f, v2 = 0.0f;
    if (hi == 0) {
      const float* s = st + ((size_t)c * CHUNKS + base + r) * 2;
      v1 = s[0]; v2 = s[1];
    }
    s1s[wave][lane] = v1;
    s2s[wave][lane] = v2;
  }
  float* yw = ybuf[wave];

#if HAVE_TDM
  tdm_load_tile16((uint32_t)(uintptr_t)&xbuf[wave][0][0],
                  (uint64_t)(uintptr_t)(xc + (size_t)base * CLEN), 16);
#endif

  for (int blk = 0; blk < NBLK; ++blk) {
    const int t0 = blk * TBLK;
    const int cur = blk & 1;
    float* xw = &xbuf[wave][cur][0];
#if HAVE_TDM
    if (blk + 1 < NBLK) {
      asm volatile("s_wait_dscnt 0x0" ::: "memory");
      tdm_load_tile16((uint32_t)(uintptr_t)&xbuf[wave][cur ^ 1][0],
                      (uint64_t)(uintptr_t)(xc + (size_t)base * CLEN + t0 + TBLK), 16);
      __builtin_amdgcn_s_wait_tensorcnt(1);
    } else {
      __builtin_amdgcn_s_wait_tensorcnt(0);
    }
    wave_sync();
#else
    #pragma unroll
    for (int k = 0; k < 8; ++k) {              // 256 floats/wave, 64B runs (L2)
      int flat = lane + 32 * k;
      int n = flat >> 4, j = flat & 15;
      xw[n * 17 + j] = xc[(size_t)(base + n) * CLEN + t0 + j];
    }
    if (blk + 1 < NBLK)
      __builtin_prefetch(&xc[(size_t)(base + r) * CLEN + t0 + TBLK], 0, 1);
    wave_sync();
#endif

    v8f acc = {0.f, 0.f, 0.f, 0.f, 0.f, 0.f, 0.f, 0.f};
#if HAVE_WMMA
    #pragma unroll
    for (int kk = 0; kk < 4; ++kk) {           // B 4x16 frag: v0=row(4kk+2*hi), v1=+1
      int rb = 4 * kk + 2 * hi;
      v2f bf;
      bf.x = xw[r * 17 + rb];
      bf.y = xw[r * 17 + rb + 1];
      acc = __builtin_amdgcn_wmma_f32_16x16x4_f32(false, afrag[kk], false, bf,
                                                  (short)0, acc, false, false);
    }
    {                                          // state cols: rows 16/17 (18/19 = 0)
      v2f bf;
      bf.x = s1s[wave][lane];                  // unconditional: upper lanes read 0
      bf.y = s2s[wave][lane];
      acc = __builtin_amdgcn_wmma_f32_16x16x4_f32(false, afrag[4], false, bf,
                                                  (short)0, acc, false, false);
    }
#else
    for (int v = 0; v < 8; ++v) {              // scalar fallback, same math
      int i = v + 8 * hi;
      float sum = fmaf(hg[16 + i], s1s[wave][r], hg[32 + i] * s2s[wave][r]);
      for (int j = 0; j <= i; ++j) sum = fmaf(hg[i - j], xw[r * 17 + j], sum);
      acc[v] = sum;
    }
#endif

#if HAVE_ASYNC_ST
    __builtin_amdgcn_s_wait_asynccnt(0);       // WAR: prior async reads of ybuf done
#endif
    #pragma unroll
    for (int v = 0; v < 8; ++v)                // D layout: v=row (+8 upper half)
      yw[r * 17 + v + 8 * hi] = acc[v];
    wave_sync();

    if (hi == 0) {                             // next-block state from last 2 (x,y)
      float x14 = xw[r * 17 + 14], x15 = xw[r * 17 + 15];
      float y14 = yw[r * 17 + 14], y15 = yw[r * 17 + 15];
      float s2m = fmaf(B2, x14, -A2 * y14);    // s2 entering step 15
      s1s[wave][lane] = fmaf(B1, x15, fmaf(-A1, y15, s2m));
      s2s[wave][lane] = fmaf(B2, x15, -A2 * y15);
    }

#if HAVE_ASYNC_ST
    asm volatile("s_wait_dscnt 0x0" ::: "memory");     // yw writes visible to async
    #pragma unroll
    for (int k = 0; k < 8; ++k) {              // LDS -> global, no VGPR roundtrip
      int flat = lane + 32 * k;
      int n = flat >> 4, j = flat & 15;
      float* gp = &yc[(size_t)(base + n) * CLEN + t0 + j];
      uint32_t lo = (uint32_t)(uintptr_t)&yw[n * 17 + j];
      __builtin_amdgcn_global_store_async_from_lds_b32(
          (glb_i32*)(uintptr_t)gp, (lds_i32*)(uintptr_t)lo, 0, 0);
    }
#else
    #pragma unroll
    for (int k = 0; k < 8; ++k) {              // coalesced Y store
      int flat = lane + 32 * k;
      int n = flat >> 4, j = flat & 15;
      yc[(size_t)(base + n) * CLEN + t0 + j] = yw[n * 17 + j];
    }
#endif
    wave_sync();
  }
}

// ---------------------------------------------------------------------------
extern "C" void kernel_launch(void* const* d_in, const int* in_sizes, int n_in,
                              void* d_out, int out_size, void* d_ws, size_t ws_size,
                              hipStream_t stream) {
  const float* x  = (const float*)d_in[0];
  const float* b0 = (const float*)d_in[1];
  const float* b1 = (const float*)d_in[2];
  const float* b2 = (const float*)d_in[3];
  const float* a1 = (const float*)d_in[4];
  const float* a2 = (const float*)d_in[5];
  float* y  = (float*)d_out;
  float* st = (float*)d_ws;                    // NCH*CHUNKS*2 floats = 128 KiB

  biquad_pass1<<<NCH, 256, 0, stream>>>(x, b0, b1, b2, a1, a2, st);
  biquad_pass2<<<(NCH * 32 + 255) / 256, 256, 0, stream>>>(a1, a2, st);
  biquad_pass3<<<NCH * 2, 256, 0, stream>>>(x, b0, b1, b2, a1, a2, st, y);
}